// FECAMLayer_36301063586381
// MI455X (gfx1250) — compile-verified
//
#include <hip/hip_runtime.h>
#include <hip/hip_bf16.h>
#include <math.h>

typedef float v2f __attribute__((ext_vector_type(2)));
typedef float v8f __attribute__((ext_vector_type(8)));

#define LL 512
#define HH 1024
#define NCOL 32
#define NTHREADS 512

__device__ __forceinline__ v8f wmma_f32(v2f a, v2f b, v8f c) {
  // D = A(16x4 f32) * B(4x16 f32) + C(16x16 f32)
  return __builtin_amdgcn_wmma_f32_16x16x4_f32(false, a, false, b, (short)0, c, false, false);
}

// 4-row interleaved LDS layout: S[idx4(k,j)] holds logical S[k][j].
// Rows k, k+1 (k even) are adjacent floats -> single aligned ds_load_b64 for
// the WMMA B operand, and lanes 0-15 (kk=0) / 16-31 (kk=2) split one 16B
// group -> conflict-free across all 64 LDS banks.
__device__ __forceinline__ int idx4(int k, int j) {
  return ((k >> 2) * NCOL + j) * 4 + (k & 3);
}

// D[k][m] = 2*cos(pi*k*(2m+1)/(2L)); integer phase reduced mod 2048 for accuracy.
__global__ void build_dct_kernel(float* __restrict__ D) {
  int i = blockIdx.x * blockDim.x + threadIdx.x;
  if (i < LL * LL) {
    int k = i >> 9;
    int m = i & (LL - 1);
    int t = (k * (2 * m + 1)) & (4 * LL - 1);  // cos period: 2048 phase units
    D[i] = 2.0f * cosf((float)t * (float)(M_PI / 1024.0));
  }
}

__device__ __forceinline__ void layernorm_cols(float* __restrict__ S,
                                               const float* __restrict__ gamma,
                                               const float* __restrict__ beta,
                                               float* __restrict__ Psum,
                                               float* __restrict__ Psq,
                                               float* __restrict__ MuS,
                                               float* __restrict__ RsS,
                                               int tid) {
  const int j  = tid & (NCOL - 1);
  const int r0 = tid >> 5;  // 0..15, uniform within a wave
  float s = 0.f, q = 0.f;
  for (int r = r0; r < LL; r += 16) {
    float v = S[idx4(r, j)];
    s += v; q += v * v;
  }
  Psum[r0 * NCOL + j] = s;
  Psq[r0 * NCOL + j]  = q;
  __syncthreads();
  if (tid < NCOL) {
    float ss = 0.f, qq = 0.f;
    for (int i = 0; i < 16; ++i) { ss += Psum[i * NCOL + tid]; qq += Psq[i * NCOL + tid]; }
    float mu  = ss * (1.0f / LL);
    float var = qq * (1.0f / LL) - mu * mu;
    MuS[tid] = mu;
    RsS[tid] = rsqrtf(var + 1e-6f);
  }
  __syncthreads();
  float mu = MuS[j], rs = RsS[j];
  for (int r = r0; r < LL; r += 16) {
    float v = S[idx4(r, j)];
    S[idx4(r, j)] = (v - mu) * rs * gamma[r] + beta[r];
  }
  __syncthreads();
}

__launch_bounds__(NTHREADS, 1)
__global__ void fecam_fused_kernel(const float* __restrict__ x,
                                   const float* __restrict__ gamma,
                                   const float* __restrict__ beta,
                                   const float* __restrict__ w1,
                                   const float* __restrict__ w2,
                                   const float* __restrict__ Dm,
                                   float* __restrict__ out) {
  __shared__ float Xs[LL * NCOL];   //  64 KB : x slice [l][c] (interleaved)
  __shared__ float Ss[LL * NCOL];   //  64 KB : freq -> sd -> fw -> fw_norm
  __shared__ float Hs[HH * NCOL];   // 128 KB : hidden relu activations
  __shared__ float Psum[16 * NCOL];
  __shared__ float Psq[16 * NCOL];
  __shared__ float MuS[NCOL];
  __shared__ float RsS[NCOL];

  const int tid  = threadIdx.x;
  const int lane = tid & 31;
  const int wave = tid >> 5;          // 0..15
  const int b    = blockIdx.x >> 4;
  const int c0   = (blockIdx.x & 15) * NCOL;

  const float* xb = x + (size_t)b * LL * LL;

  // ---- Load X slice: Xs[l][j] = x[b, l, c0+j]
  for (int idx = tid; idx < LL * NCOL; idx += NTHREADS) {
    int l = idx >> 5, j = idx & (NCOL - 1);
    Xs[idx4(l, j)] = xb[l * LL + c0 + j];
  }
  __syncthreads();

  const int arow_lane = lane & 15;
  const int kk        = (lane >> 4) * 2;   // 0 or 2 (low bit provably 0)
  const int coln      = lane & 15;

  // ================= GEMM1: F = D @ Xs  (M=512, K=512, N=32) =================
  {
    const int mt0 = wave, mt1 = wave + 16;
    v8f acc00 = {}, acc01 = {}, acc10 = {}, acc11 = {};
    const float* a0p = Dm + (mt0 * 16 + arow_lane) * LL + kk;
    const float* a1p = Dm + (mt1 * 16 + arow_lane) * LL + kk;
    for (int k0 = 0; k0 < LL; k0 += 4) {
      v2f a0 = *(const v2f*)(a0p + k0);
      v2f a1 = *(const v2f*)(a1p + k0);
      int kb = k0 + kk;                      // kb & 3 == kk
      v2f b0 = *(const v2f*)(&Xs[idx4(kb, coln)]);
      v2f b1 = *(const v2f*)(&Xs[idx4(kb, coln + 16)]);
      acc00 = wmma_f32(a0, b0, acc00);
      acc01 = wmma_f32(a0, b1, acc01);
      acc10 = wmma_f32(a1, b0, acc10);
      acc11 = wmma_f32(a1, b1, acc11);
    }
    const int rbase = (lane >> 4) * 8;
    for (int i = 0; i < 8; ++i) {
      int r0 = mt0 * 16 + rbase + i;
      int r1 = mt1 * 16 + rbase + i;
      Ss[idx4(r0, coln)]      = acc00[i];
      Ss[idx4(r0, coln + 16)] = acc01[i];
      Ss[idx4(r1, coln)]      = acc10[i];
      Ss[idx4(r1, coln + 16)] = acc11[i];
    }
  }
  __syncthreads();

  // ---- LayerNorm #1 over k (rows), per column
  layernorm_cols(Ss, gamma, beta, Psum, Psq, MuS, RsS, tid);

  // ================= GEMM2: H = relu(W1 @ SD)  (M=1024, K=512, N=32) =========
  {
    v8f acc[4][2];
    #pragma unroll
    for (int i = 0; i < 4; ++i) { acc[i][0] = (v8f){}; acc[i][1] = (v8f){}; }
    for (int k0 = 0; k0 < LL; k0 += 4) {
      int kb = k0 + kk;
      v2f b0 = *(const v2f*)(&Ss[idx4(kb, coln)]);
      v2f b1 = *(const v2f*)(&Ss[idx4(kb, coln + 16)]);
      #pragma unroll
      for (int i = 0; i < 4; ++i) {
        int arow = (wave + 16 * i) * 16 + arow_lane;
        v2f a = *(const v2f*)(w1 + arow * LL + k0 + kk);
        acc[i][0] = wmma_f32(a, b0, acc[i][0]);
        acc[i][1] = wmma_f32(a, b1, acc[i][1]);
      }
    }
    const int rbase = (lane >> 4) * 8;
    #pragma unroll
    for (int i = 0; i < 4; ++i)
      for (int e = 0; e < 8; ++e) {
        int r = (wave + 16 * i) * 16 + rbase + e;
        Hs[idx4(r, coln)]      = fmaxf(acc[i][0][e], 0.f);
        Hs[idx4(r, coln + 16)] = fmaxf(acc[i][1][e], 0.f);
      }
  }
  __syncthreads();

  // ================= GEMM3: FW = sigmoid(W2 @ H)  (M=512, K=1024, N=32) ======
  {
    const int mt0 = wave, mt1 = wave + 16;
    v8f acc00 = {}, acc01 = {}, acc10 = {}, acc11 = {};
    const float* a0p = w2 + (mt0 * 16 + arow_lane) * HH + kk;
    const float* a1p = w2 + (mt1 * 16 + arow_lane) * HH + kk;
    for (int k0 = 0; k0 < HH; k0 += 4) {
      v2f a0 = *(const v2f*)(a0p + k0);
      v2f a1 = *(const v2f*)(a1p + k0);
      int kb = k0 + kk;
      v2f b0 = *(const v2f*)(&Hs[idx4(kb, coln)]);
      v2f b1 = *(const v2f*)(&Hs[idx4(kb, coln + 16)]);
      acc00 = wmma_f32(a0, b0, acc00);
      acc01 = wmma_f32(a0, b1, acc01);
      acc10 = wmma_f32(a1, b0, acc10);
      acc11 = wmma_f32(a1, b1, acc11);
    }
    const int rbase = (lane >> 4) * 8;
    for (int i = 0; i < 8; ++i) {
      int r0 = mt0 * 16 + rbase + i;
      int r1 = mt1 * 16 + rbase + i;
      Ss[idx4(r0, coln)]      = 1.0f / (1.0f + expf(-acc00[i]));
      Ss[idx4(r0, coln + 16)] = 1.0f / (1.0f + expf(-acc01[i]));
      Ss[idx4(r1, coln)]      = 1.0f / (1.0f + expf(-acc10[i]));
      Ss[idx4(r1, coln + 16)] = 1.0f / (1.0f + expf(-acc11[i]));
    }
  }
  __syncthreads();

  // ---- LayerNorm #2 (same gamma/beta)
  layernorm_cols(Ss, gamma, beta, Psum, Psq, MuS, RsS, tid);

  // ---- out[b, l, c0+j] = X[l][j] * FWnorm[l][j]
  float* ob = out + (size_t)b * LL * LL;
  for (int idx = tid; idx < LL * NCOL; idx += NTHREADS) {
    int l = idx >> 5, j = idx & (NCOL - 1);
    ob[l * LL + c0 + j] = Xs[idx4(l, j)] * Ss[idx4(l, j)];
  }
}

extern "C" void kernel_launch(void* const* d_in, const int* in_sizes, int n_in,
                              void* d_out, int out_size, void* d_ws, size_t ws_size,
                              hipStream_t stream) {
  const float* x     = (const float*)d_in[0];
  const float* gamma = (const float*)d_in[1];
  const float* beta  = (const float*)d_in[2];
  const float* w1    = (const float*)d_in[3];
  const float* w2    = (const float*)d_in[4];
  float* out = (float*)d_out;
  float* Dm  = (float*)d_ws;  // 512*512 f32 = 1 MB scratch for the DCT basis

  const int nbatch = in_sizes[0] / (LL * LL);  // 128

  build_dct_kernel<<<(LL * LL + 255) / 256, 256, 0, stream>>>(Dm);
  fecam_fused_kernel<<<nbatch * (LL / NCOL), NTHREADS, 0, stream>>>(
      x, gamma, beta, w1, w2, Dm, out);
}